// DeepMemoryUnit_60447369724367
// MI455X (gfx1250) — compile-verified
//
#include <hip/hip_runtime.h>

// ---------------------------------------------------------------------------
// DeepMemoryUnit for MI455X (gfx1250, wave32).
//
// Roofline: memories = 134MB fp32; 2 reads + 1 write => ~400MB HBM traffic
// => ~17us floor @ 23.3 TB/s. Attention math ~270 MFLOP (negligible).
// The only dense-matrix work is the 4 bank-routed linears; we compute
// Z[j] = X @ W[j] for all 16 banks with V_WMMA_F32_16X16X4_F32 (exact fp32),
// then blend the top-2 selected banks per batch row. Weight tensors are read
// once each (unavoidable), attention kernels are pure coalesced streaming.
// ---------------------------------------------------------------------------

#define BATCHN  32
#define DMODEL  1024
#define NHEADS  8
#define DMEM    64
#define BANKS   16
#define MSIZE   2048
#define TOPK    2
#define HDIM    512   // NHEADS * DMEM

typedef float v2f __attribute__((ext_vector_type(2)));
typedef float v8f __attribute__((ext_vector_type(8)));

// ---------------------------------------------------------------------------
// Bank GEMM: Z[j] = X @ W[j].  X:(32,K)  W:(BANKS,K,N)  Z:(BANKS,32,N)
// Grid: (N/128, 32/16, BANKS). Block: 256 = 8 waves; each wave owns a
// 16x16 output tile (wave w covers columns nb+16w .. nb+16w+15).
// WMMA f32 16x16x4: A = X tile (16 rows x 4 k), B = W (4 k x 16 cols).
// A-layout (ISA 7.12.2): lane l -> row = l&15, khalf = l>>4,
//   vgpr0 = K(2*khalf), vgpr1 = K(2*khalf+1)  -> contiguous float2 from LDS.
// ---------------------------------------------------------------------------
template<int KDIM, int NDIM>
__global__ __launch_bounds__(256) void bank_gemm(
    const float* __restrict__ X, const float* __restrict__ W,
    float* __restrict__ Z)
{
  const int tid   = threadIdx.x;
  const int lane  = tid & 31;
  const int wv    = tid >> 5;                 // wave id, 0..7 (wave32)
  const int j     = blockIdx.z;               // bank
  const int mb    = blockIdx.y * 16;          // batch-row tile base
  const int nb    = blockIdx.x * 128 + wv * 16;

  const int row   = lane & 15;                // A row / D column-lane
  const int khalf = lane >> 4;                // 0 or 1
  const int col   = nb + row;                 // B/D column

  __shared__ float sX[16 * 257];              // 16 x 256 K-chunk, pad vs banks

  const float* __restrict__ Wj = W + (size_t)j * KDIM * NDIM;
  v8f acc = {};

  for (int kb = 0; kb < KDIM; kb += 256) {
    // cooperative, coalesced stage of X tile (16 rows x 256 k)
    #pragma unroll
    for (int i = 0; i < 16; ++i) {
      int lin = i * 256 + tid;
      int r = lin >> 8, c = lin & 255;
      sX[r * 257 + c] = X[(size_t)(mb + r) * KDIM + kb + c];
    }
    __syncthreads();

    #pragma unroll 4
    for (int kc = 0; kc < 256; kc += 4) {
      const int kk = kc + 2 * khalf;
      v2f a, b;
      a.x = sX[row * 257 + kk];
      a.y = sX[row * 257 + kk + 1];
      const float* wp = Wj + (size_t)(kb + kk) * NDIM + col;
      b.x = wp[0];
      b.y = wp[NDIM];
      // D = A x B + C, fp32 matrix core (16x16x4)
      acc = __builtin_amdgcn_wmma_f32_16x16x4_f32(
          false, a, false, b, (short)0, acc, false, false);
    }
    __syncthreads();
  }

  // D layout: vgpr r -> M = r + 8*(lane>>4), N = lane&15
  #pragma unroll
  for (int r = 0; r < 8; ++r) {
    int m = mb + r + 8 * khalf;
    Z[((size_t)j * BATCHN + m) * NDIM + col] = acc[r];
  }
}

// ---------------------------------------------------------------------------
// Blend the top-k selected banks:  Y[b,o] = sum_k p[b,k]*(Z[sel[b,k],b,o] + bias[sel[b,k],o])
// ---------------------------------------------------------------------------
__global__ __launch_bounds__(256) void blend_banks(
    const float* __restrict__ Z, const float* __restrict__ bias,
    const int* __restrict__ sel, const float* __restrict__ probs,
    float* __restrict__ Y, int NDIM)
{
  int idx = blockIdx.x * 256 + threadIdx.x;
  int total = BATCHN * NDIM;
  if (idx >= total) return;
  int b = idx / NDIM, o = idx - b * NDIM;
  float y = 0.f;
  #pragma unroll
  for (int k = 0; k < TOPK; ++k) {
    int s = sel[b * TOPK + k];
    float p = probs[b * TOPK + k];
    y += p * (Z[((size_t)s * BATCHN + b) * NDIM + o] + bias[(size_t)s * NDIM + o]);
  }
  Y[idx] = y;
}

// ---------------------------------------------------------------------------
// express_memory: per (b,h) block: softmax(q . mem / 8) @ mem  -> read_heads
// Streaming-bound: 512KB fp32 tile per block, coalesced b128 loads.
// ---------------------------------------------------------------------------
__global__ __launch_bounds__(256) void attend_read(
    const float* __restrict__ q_heads, const float* __restrict__ memories,
    float* __restrict__ read_heads)
{
  const int bh  = blockIdx.x;               // b*NHEADS + h
  const int tid = threadIdx.x;
  const float* __restrict__ mem = memories + (size_t)bh * MSIZE * DMEM;

  __shared__ float sq[DMEM];
  __shared__ float sp[MSIZE];
  __shared__ float sred[256];

  if (tid < DMEM) sq[tid] = q_heads[bh * DMEM + tid];
  __syncthreads();

  // phase 1: scores + running max
  float lmax = -3.402823466e38f;
  #pragma unroll
  for (int i = 0; i < MSIZE / 256; ++i) {
    int m = tid + i * 256;
    if (i + 1 < MSIZE / 256)
      __builtin_prefetch(mem + (size_t)(m + 256) * DMEM, 0, 1);
    const float4* mp = (const float4*)(mem + (size_t)m * DMEM);
    float s = 0.f;
    #pragma unroll
    for (int d4 = 0; d4 < DMEM / 4; ++d4) {
      float4 v = mp[d4];
      s += v.x * sq[4*d4] + v.y * sq[4*d4+1] + v.z * sq[4*d4+2] + v.w * sq[4*d4+3];
    }
    s *= 0.125f;                             // 1/sqrt(64)
    sp[m] = s;
    lmax = fmaxf(lmax, s);
  }
  sred[tid] = lmax; __syncthreads();
  for (int off = 128; off > 0; off >>= 1) {
    if (tid < off) sred[tid] = fmaxf(sred[tid], sred[tid + off]);
    __syncthreads();
  }
  const float gmax = sred[0]; __syncthreads();

  float lsum = 0.f;
  #pragma unroll
  for (int i = 0; i < MSIZE / 256; ++i) {
    int m = tid + i * 256;
    float e = __expf(sp[m] - gmax);
    sp[m] = e; lsum += e;
  }
  sred[tid] = lsum; __syncthreads();
  for (int off = 128; off > 0; off >>= 1) {
    if (tid < off) sred[tid] += sred[tid + off];
    __syncthreads();
  }
  const float inv = 1.0f / sred[0]; __syncthreads();
  #pragma unroll
  for (int i = 0; i < MSIZE / 256; ++i) sp[tid + i * 256] *= inv;
  __syncthreads();

  // phase 2: read[d] = sum_m p[m] * mem[m,d]  (d = tid&63 -> coalesced rows)
  const int d  = tid & 63;
  const int rg = tid >> 6;                   // 4 row-groups
  float acc = 0.f;
  for (int m = rg; m < MSIZE; m += 4)
    acc += sp[m] * mem[(size_t)m * DMEM + d];
  sred[tid] = acc; __syncthreads();
  if (tid < DMEM)
    read_heads[bh * DMEM + tid] =
        sred[tid] + sred[tid + 64] + sred[tid + 128] + sred[tid + 192];
}

// ---------------------------------------------------------------------------
// store_memory: per (b,h): w = softmax(wq . mem / 8);
//               new_mem = mem + w (outer) st     (streaming read+write)
// ---------------------------------------------------------------------------
__global__ __launch_bounds__(256) void attend_store(
    const float* __restrict__ wq_heads, const float* __restrict__ st_heads,
    const float* __restrict__ memories, float* __restrict__ new_memories)
{
  const int bh  = blockIdx.x;
  const int tid = threadIdx.x;
  const float* __restrict__ mem = memories + (size_t)bh * MSIZE * DMEM;
  float* __restrict__ out = new_memories + (size_t)bh * MSIZE * DMEM;

  __shared__ float sq[DMEM];
  __shared__ float sst[DMEM];
  __shared__ float sp[MSIZE];
  __shared__ float sred[256];

  if (tid < DMEM) {
    sq[tid]  = wq_heads[bh * DMEM + tid];
    sst[tid] = st_heads[bh * DMEM + tid];
  }
  __syncthreads();

  float lmax = -3.402823466e38f;
  #pragma unroll
  for (int i = 0; i < MSIZE / 256; ++i) {
    int m = tid + i * 256;
    if (i + 1 < MSIZE / 256)
      __builtin_prefetch(mem + (size_t)(m + 256) * DMEM, 0, 1);
    const float4* mp = (const float4*)(mem + (size_t)m * DMEM);
    float s = 0.f;
    #pragma unroll
    for (int d4 = 0; d4 < DMEM / 4; ++d4) {
      float4 v = mp[d4];
      s += v.x * sq[4*d4] + v.y * sq[4*d4+1] + v.z * sq[4*d4+2] + v.w * sq[4*d4+3];
    }
    s *= 0.125f;
    sp[m] = s;
    lmax = fmaxf(lmax, s);
  }
  sred[tid] = lmax; __syncthreads();
  for (int off = 128; off > 0; off >>= 1) {
    if (tid < off) sred[tid] = fmaxf(sred[tid], sred[tid + off]);
    __syncthreads();
  }
  const float gmax = sred[0]; __syncthreads();

  float lsum = 0.f;
  #pragma unroll
  for (int i = 0; i < MSIZE / 256; ++i) {
    int m = tid + i * 256;
    float e = __expf(sp[m] - gmax);
    sp[m] = e; lsum += e;
  }
  sred[tid] = lsum; __syncthreads();
  for (int off = 128; off > 0; off >>= 1) {
    if (tid < off) sred[tid] += sred[tid + off];
    __syncthreads();
  }
  const float inv = 1.0f / sred[0]; __syncthreads();
  #pragma unroll
  for (int i = 0; i < MSIZE / 256; ++i) sp[tid + i * 256] *= inv;
  __syncthreads();

  // phase 2: flat elementwise update, fully coalesced
  for (int i = 0; i < (MSIZE * DMEM) / 256; ++i) {
    int idx = tid + i * 256;
    int m = idx >> 6, d = idx & 63;
    out[idx] = mem[idx] + sp[m] * sst[d];
  }
}

// ---------------------------------------------------------------------------
// response = layernorm(query + read) * scale + bias; one block per row.
// ---------------------------------------------------------------------------
__global__ __launch_bounds__(256) void residual_layernorm(
    const float* __restrict__ query, const float* __restrict__ readv,
    const float* __restrict__ scale, const float* __restrict__ bias,
    float* __restrict__ out)
{
  const int b = blockIdx.x, tid = threadIdx.x;
  __shared__ float sred[256];

  float v[DMODEL / 256];
  float lsum = 0.f;
  #pragma unroll
  for (int i = 0; i < DMODEL / 256; ++i) {
    int c = tid + i * 256;
    v[i] = query[(size_t)b * DMODEL + c] + readv[(size_t)b * DMODEL + c];
    lsum += v[i];
  }
  sred[tid] = lsum; __syncthreads();
  for (int off = 128; off > 0; off >>= 1) {
    if (tid < off) sred[tid] += sred[tid + off];
    __syncthreads();
  }
  const float mu = sred[0] * (1.0f / DMODEL); __syncthreads();

  float lvar = 0.f;
  #pragma unroll
  for (int i = 0; i < DMODEL / 256; ++i) { float d = v[i] - mu; lvar += d * d; }
  sred[tid] = lvar; __syncthreads();
  for (int off = 128; off > 0; off >>= 1) {
    if (tid < off) sred[tid] += sred[tid + off];
    __syncthreads();
  }
  const float rstd = rsqrtf(sred[0] * (1.0f / DMODEL) + 1e-5f);

  #pragma unroll
  for (int i = 0; i < DMODEL / 256; ++i) {
    int c = tid + i * 256;
    out[(size_t)b * DMODEL + c] = (v[i] - mu) * rstd * scale[c] + bias[c];
  }
}

// ---------------------------------------------------------------------------
extern "C" void kernel_launch(void* const* d_in, const int* in_sizes, int n_in,
                              void* d_out, int out_size, void* d_ws, size_t ws_size,
                              hipStream_t stream)
{
  const float* query    = (const float*)d_in[0];
  const int*   sel      = (const int*)  d_in[1];
  const float* probs    = (const float*)d_in[2];
  const float* memories = (const float*)d_in[3];
  const float* W_read   = (const float*)d_in[4];
  const float* b_read   = (const float*)d_in[5];
  const float* W_merge  = (const float*)d_in[6];
  const float* b_merge  = (const float*)d_in[7];
  const float* W_wq     = (const float*)d_in[8];
  const float* b_wq     = (const float*)d_in[9];
  const float* W_ws     = (const float*)d_in[10];
  const float* b_ws     = (const float*)d_in[11];
  const float* ln_s     = (const float*)d_in[12];
  const float* ln_b     = (const float*)d_in[13];

  float* response = (float*)d_out;                       // (B, DMODEL)
  float* new_mem  = (float*)d_out + BATCHN * DMODEL;     // (B,H,M,Dm)

  // workspace layout (floats); Z reused serially between GEMM+blend pairs
  float* Z          = (float*)d_ws;                              // 16*32*1024
  float* q_heads    = Z + (size_t)BANKS * BATCHN * DMODEL;       // 32*512
  float* wq_heads   = q_heads    + BATCHN * HDIM;
  float* read_heads = wq_heads   + BATCHN * HDIM;
  float* st_heads   = read_heads + BATCHN * HDIM;
  float* readv      = st_heads   + BATCHN * HDIM;                // 32*1024

  const dim3 blk(256);
  const dim3 g512 (HDIM   / 128, BATCHN / 16, BANKS);  // (4,2,16)
  const dim3 g1024(DMODEL / 128, BATCHN / 16, BANKS);  // (8,2,16)
  const int nb512  = (BATCHN * HDIM   + 255) / 256;
  const int nb1024 = (BATCHN * DMODEL + 255) / 256;

  // q_heads = vlinear(query, W_read)
  bank_gemm<DMODEL, HDIM><<<g512, blk, 0, stream>>>(query, W_read, Z);
  blend_banks<<<nb512, blk, 0, stream>>>(Z, b_read, sel, probs, q_heads, HDIM);

  // wq_heads = vlinear(query, W_wq)
  bank_gemm<DMODEL, HDIM><<<g512, blk, 0, stream>>>(query, W_wq, Z);
  blend_banks<<<nb512, blk, 0, stream>>>(Z, b_wq, sel, probs, wq_heads, HDIM);

  // read_heads = express_memory(q_heads, memories)
  attend_read<<<BATCHN * NHEADS, blk, 0, stream>>>(q_heads, memories, read_heads);

  // read = vlinear(read_heads, W_merge)
  bank_gemm<HDIM, DMODEL><<<g1024, blk, 0, stream>>>(read_heads, W_merge, Z);
  blend_banks<<<nb1024, blk, 0, stream>>>(Z, b_merge, sel, probs, readv, DMODEL);

  // response = layernorm(query + read)
  residual_layernorm<<<BATCHN, blk, 0, stream>>>(query, readv, ln_s, ln_b, response);

  // st_heads = vlinear(response, W_ws)
  bank_gemm<DMODEL, HDIM><<<g512, blk, 0, stream>>>(response, W_ws, Z);
  blend_banks<<<nb512, blk, 0, stream>>>(Z, b_ws, sel, probs, st_heads, HDIM);

  // new_memories = store_memory(wq_heads, st_heads, memories)
  attend_store<<<BATCHN * NHEADS, blk, 0, stream>>>(wq_heads, st_heads, memories, new_mem);
}